// MeanRecallLoss_12687333392526
// MI455X (gfx1250) — compile-verified
//
#include <hip/hip_runtime.h>
#include <hip/hip_bf16.h>
#include <stdint.h>

#define NUM_CLASSES 1000
#define BATCH       131072
#define WPB         4          // waves per block
#define TPB         (WPB * 32) // threads per block
#define GRID        4096       // 4096 blocks * 4 waves = 16384 waves -> 8 rows/wave

typedef __attribute__((address_space(3))) float lds_f;

// ---------------------------------------------------------------------------
// Kernel 0: zero the workspace histograms (counts[1000], hits[1000])
// ---------------------------------------------------------------------------
__global__ void mrl_zero_kernel(unsigned* __restrict__ ws) {
    int i = blockIdx.x * blockDim.x + threadIdx.x;
    if (i < 2 * NUM_CLASSES) ws[i] = 0u;
}

// ---------------------------------------------------------------------------
// Kernel 1: per-row argmax of pred & true, histogram by true class.
// One wave per row; async global->LDS double buffering (gfx1250 ASYNCcnt path).
// ---------------------------------------------------------------------------
__global__ __launch_bounds__(TPB) void mrl_argmax_kernel(
        const float* __restrict__ pred,
        const float* __restrict__ tru,
        unsigned* __restrict__ counts,
        unsigned* __restrict__ hits) {

    // Per-wave LDS: 4096 floats = 16KB: [bufP0 1024][bufT0 1024][bufP1 1024][bufT1 1024]
    __shared__ __align__(16) float smem[WPB * 4096];

    const int lane = threadIdx.x & 31;
    const int wave = threadIdx.x >> 5;
    const unsigned gw     = blockIdx.x * WPB + wave;
    const unsigned stride = gridDim.x * WPB;

    float* wbuf = &smem[wave * 4096];
    // 32-bit LDS byte offset of this wave's buffer (for the async-to-LDS ops)
    const unsigned wbase   = (unsigned)(uintptr_t)(lds_f*)wbuf;
    const unsigned laneOff = (unsigned)lane * 16u;

    const uint64_t totalBytes = (uint64_t)BATCH * NUM_CLASSES * 4ull;

    // Fast issue: full 1024-float tiles for pred+true, no bounds checks.
    // INST_OFFSET is added to BOTH the LDS address and the global address
    // (CDNA5 ISA), so one base VGPR (pair) per matrix suffices.
    auto issue_fast = [&](unsigned row, int b) {
        const uint64_t rowByte = (uint64_t)row * (NUM_CLASSES * 4ull);
        const uint64_t gp = (uint64_t)(uintptr_t)pred + rowByte + laneOff;
        const uint64_t gt = (uint64_t)(uintptr_t)tru  + rowByte + laneOff;
        const unsigned lp = wbase + (unsigned)b * 8192u + laneOff;
        const unsigned lt = lp + 4096u;
        asm volatile(
            "global_load_async_to_lds_b128 %0, %2, off\n\t"
            "global_load_async_to_lds_b128 %0, %2, off offset:512\n\t"
            "global_load_async_to_lds_b128 %0, %2, off offset:1024\n\t"
            "global_load_async_to_lds_b128 %0, %2, off offset:1536\n\t"
            "global_load_async_to_lds_b128 %0, %2, off offset:2048\n\t"
            "global_load_async_to_lds_b128 %0, %2, off offset:2560\n\t"
            "global_load_async_to_lds_b128 %0, %2, off offset:3072\n\t"
            "global_load_async_to_lds_b128 %0, %2, off offset:3584\n\t"
            "global_load_async_to_lds_b128 %1, %3, off\n\t"
            "global_load_async_to_lds_b128 %1, %3, off offset:512\n\t"
            "global_load_async_to_lds_b128 %1, %3, off offset:1024\n\t"
            "global_load_async_to_lds_b128 %1, %3, off offset:1536\n\t"
            "global_load_async_to_lds_b128 %1, %3, off offset:2048\n\t"
            "global_load_async_to_lds_b128 %1, %3, off offset:2560\n\t"
            "global_load_async_to_lds_b128 %1, %3, off offset:3072\n\t"
            "global_load_async_to_lds_b128 %1, %3, off offset:3584"
            :: "v"(lp), "v"(lt), "v"(gp), "v"(gt) : "memory");
    };

    // Safe issue (only for the very last row): per-lane EXEC-masked so the
    // 24-float tail overread past the end of the allocation never happens.
    // Still issues exactly 16 wave-level async ops (lanes 0..25 stay active
    // on the tail chunk), so ASYNCcnt accounting matches the fast path.
    auto issue_safe = [&](unsigned row, int b) {
        const uint64_t rowByte = (uint64_t)row * (NUM_CLASSES * 4ull);
        const uint64_t pAddr = (uint64_t)(uintptr_t)pred + rowByte;
        const uint64_t tAddr = (uint64_t)(uintptr_t)tru  + rowByte;
        const unsigned pLds = wbase + (unsigned)b * 8192u;
        const unsigned tLds = pLds + 4096u;
#pragma unroll
        for (int j = 0; j < 8; ++j) {
            const unsigned byteOff = (unsigned)j * 512u + laneOff;
            const bool ok = (rowByte + byteOff + 16ull) <= totalBytes;
            if (ok) {
                unsigned lp = pLds + byteOff;
                uint64_t gp = pAddr + byteOff;
                asm volatile("global_load_async_to_lds_b128 %0, %1, off"
                             :: "v"(lp), "v"(gp) : "memory");
                unsigned lt = tLds + byteOff;
                uint64_t gt = tAddr + byteOff;
                asm volatile("global_load_async_to_lds_b128 %0, %1, off"
                             :: "v"(lt), "v"(gt) : "memory");
            }
        }
    };

    auto issue_tile = [&](unsigned row, int b) {
        if (row != (unsigned)(BATCH - 1)) issue_fast(row, b);
        else                              issue_safe(row, b);
    };

    // Consume buffer b: per-lane argmax over 32 elements, wave reduce, atomics.
    auto compute_tile = [&](int b) {
        const float4* p4 = (const float4*)(wbuf + b * 2048);
        const float4* t4 = (const float4*)(wbuf + b * 2048 + 1024);
        float pv = -INFINITY, tv = -INFINITY;
        int   pi = 0,          ti = 0;
        // chunks 0..6: every idx < 896 < NUM_CLASSES, no masking needed
#pragma unroll
        for (int j = 0; j < 7; ++j) {
            float4 a = p4[j * 32 + lane];
            float4 c = t4[j * 32 + lane];
            const int base = j * 128 + lane * 4;
#pragma unroll
            for (int e = 0; e < 4; ++e) {
                const int idx = base + e;
                const float av = (&a.x)[e];
                const float cv = (&c.x)[e];
                if (av > pv || (av == pv && idx < pi)) { pv = av; pi = idx; }
                if (cv > tv || (cv == tv && idx < ti)) { tv = cv; ti = idx; }
            }
        }
        // chunk 7: idx in [896, 1023], mask idx >= NUM_CLASSES
        {
            float4 a = p4[7 * 32 + lane];
            float4 c = t4[7 * 32 + lane];
            const int base = 7 * 128 + lane * 4;
#pragma unroll
            for (int e = 0; e < 4; ++e) {
                const int idx = base + e;
                if (idx < NUM_CLASSES) {
                    const float av = (&a.x)[e];
                    const float cv = (&c.x)[e];
                    if (av > pv || (av == pv && idx < pi)) { pv = av; pi = idx; }
                    if (cv > tv || (cv == tv && idx < ti)) { tv = cv; ti = idx; }
                }
            }
        }
        // wave32 argmax reduction (first-index tie-break, matching jnp.argmax)
#pragma unroll
        for (int off = 16; off > 0; off >>= 1) {
            float opv = __shfl_down(pv, off, 32);
            int   opi = __shfl_down(pi, off, 32);
            float otv = __shfl_down(tv, off, 32);
            int   oti = __shfl_down(ti, off, 32);
            if (opv > pv || (opv == pv && opi < pi)) { pv = opv; pi = opi; }
            if (otv > tv || (otv == tv && oti < ti)) { tv = otv; ti = oti; }
        }
        if (lane == 0) {
            atomicAdd(&counts[ti], 1u);
            if (pi == ti) atomicAdd(&hits[ti], 1u);
        }
    };

    unsigned row = gw;
    if (row >= BATCH) return;

    int buf = 0;
    issue_tile(row, buf);                       // prologue
    while (row < BATCH) {
        const unsigned nrow = row + stride;
        if (nrow < BATCH) {
            issue_tile(nrow, buf ^ 1);          // overlap next tile's loads
            // allow only the 16 just-issued ops to remain outstanding
            asm volatile("s_wait_asynccnt 16" ::: "memory");
        } else {
            asm volatile("s_wait_asynccnt 0" ::: "memory");
        }
        compute_tile(buf);
        buf ^= 1;
        row = nrow;
    }
}

// ---------------------------------------------------------------------------
// Kernel 2: recalls -> -mean over present classes (single block)
// ---------------------------------------------------------------------------
__global__ __launch_bounds__(256) void mrl_finalize_kernel(
        const unsigned* __restrict__ counts,
        const unsigned* __restrict__ hits,
        float* __restrict__ out) {
    __shared__ float sS[256];
    __shared__ float sN[256];
    float s = 0.f, n = 0.f;
    for (int c = threadIdx.x; c < NUM_CLASSES; c += 256) {
        const unsigned cnt = counts[c];
        if (cnt > 0u) {
            s += (float)hits[c] / (float)cnt;
            n += 1.f;
        }
    }
    sS[threadIdx.x] = s;
    sN[threadIdx.x] = n;
    __syncthreads();
#pragma unroll
    for (int o = 128; o > 0; o >>= 1) {
        if (threadIdx.x < o) {
            sS[threadIdx.x] += sS[threadIdx.x + o];
            sN[threadIdx.x] += sN[threadIdx.x + o];
        }
        __syncthreads();
    }
    if (threadIdx.x == 0) {
        out[0] = (sN[0] > 0.f) ? -(sS[0] / sN[0]) : 0.f;
    }
}

// ---------------------------------------------------------------------------
extern "C" void kernel_launch(void* const* d_in, const int* in_sizes, int n_in,
                              void* d_out, int out_size, void* d_ws, size_t ws_size,
                              hipStream_t stream) {
    const float* y_pred = (const float*)d_in[0];
    const float* y_true = (const float*)d_in[1];
    float* out = (float*)d_out;

    unsigned* counts = (unsigned*)d_ws;
    unsigned* hitsp  = counts + NUM_CLASSES;

    mrl_zero_kernel<<<(2 * NUM_CLASSES + 255) / 256, 256, 0, stream>>>(counts);
    mrl_argmax_kernel<<<GRID, TPB, 0, stream>>>(y_pred, y_true, counts, hitsp);
    mrl_finalize_kernel<<<1, 256, 0, stream>>>(counts, hitsp, out);
}